// TransformerEmbedding_5007931867395
// MI455X (gfx1250) — compile-verified
//
#include <hip/hip_runtime.h>

// Problem constants from the reference: B=16, S=4096, D=512, V=22
#define Bv 16
#define Sv 4096
#define Dv 512
#define Vv 22

// Native clang vector types (usable with nontemporal/async builtins,
// unlike HIP_vector_type classes).
typedef float v4f __attribute__((ext_vector_type(4)));
typedef float v2f __attribute__((ext_vector_type(2)));
typedef int   v4i __attribute__((ext_vector_type(4)));

// Address-space-qualified pointers for the async global->LDS builtin:
// param 0: AS(1) v4i*  (global source), param 1: AS(3) v4i* (LDS dest).
typedef __attribute__((address_space(1))) v4i gv4i;
typedef __attribute__((address_space(3))) v4i lv4i;

#if defined(__gfx1250__) && __has_builtin(__builtin_amdgcn_global_load_async_to_lds_b128)
#define USE_ASYNC_LDS 1
#else
#define USE_ASYNC_LDS 0
#endif

// One block per sequence position s. 128 threads (4 waves, wave32); each lane
// owns one float4 (16 B) of the 512-float row. pe[s] is staged to LDS once and
// reused for all 16 batch rows -> pe traffic amortized 16x.
__global__ __launch_bounds__(128, 4)
void embed_main_kernel(const int*   __restrict__ x,
                       const float* __restrict__ tok_w,   // (V, D)
                       const float* __restrict__ pe,      // (S, D)
                       const float* __restrict__ probs,   // (V, S)
                       const float* __restrict__ dip,     // (V, V)
                       float*       __restrict__ out)     // (B, S, D)
{
    __shared__ float pe_s[Dv];        // 2 KB staged pe row
    __shared__ float coeff_s[Bv];     // masked (probs + dip) per batch row
    __shared__ int   xv_s[Bv];        // token id per batch row

    const int s   = blockIdx.x;
    const int tid = threadIdx.x;

    // Warm the 45 KB token-embedding table into the cache hierarchy
    // (emits global_prefetch_b8 on gfx1250).
    __builtin_prefetch((const char*)tok_w + tid * 352, 0, 3);

    // ---- Stage pe[s][0..511] into LDS ----
#if USE_ASYNC_LDS
    {
        gv4i* g = (gv4i*)((const char*)(pe + (size_t)s * Dv) + tid * 16);
        lv4i* l = (lv4i*)((char*)pe_s + tid * 16);
        // async DMA: 16 B per lane, tracked by ASYNCcnt
        __builtin_amdgcn_global_load_async_to_lds_b128(g, l, 0, 0);
    }
#else
    {
        const v4f v = ((const v4f*)(pe + (size_t)s * Dv))[tid];
        ((v4f*)pe_s)[tid] = v;
    }
#endif

    // ---- 16 lanes compute the per-(b,s) scalar coefficient in parallel ----
    if (tid < Bv) {
        const int b  = tid;
        const int xv = x[b * Sv + s];
        float c = probs[xv * Sv + s];
        if (s < Sv - 1) {
            const int xn = x[b * Sv + s + 1];
            c += dip[xv * Vv + xn];
        }
        // Fold the (x != 0) mask into the coefficient: pe_masked*(p+d) == pe*cm
        coeff_s[b] = (xv != 0) ? c : 0.0f;
        xv_s[b]    = xv;
    }

#if USE_ASYNC_LDS
  #if __has_builtin(__builtin_amdgcn_s_wait_asynccnt)
    __builtin_amdgcn_s_wait_asynccnt(0);
  #else
    asm volatile("s_wait_asynccnt 0" ::: "memory");
  #endif
#endif
    __syncthreads();

    const v4f p = ((const v4f*)pe_s)[tid];

    // ---- Hot loop: 16 batch rows, pure b128 load + v_fma + b128 NT store ----
    #pragma unroll
    for (int b = 0; b < Bv; ++b) {
        const int   xv = xv_s[b];
        const float c  = coeff_s[b];
        const v4f   t  = ((const v4f*)(tok_w + (size_t)xv * Dv))[tid];
        v4f r;
        r.x = fmaf(p.x, c, t.x);
        r.y = fmaf(p.y, c, t.y);
        r.z = fmaf(p.z, c, t.z);
        r.w = fmaf(p.w, c, t.w);
        // 128 MB streaming output, never re-read on device -> non-temporal
        v4f* dst = (v4f*)(out + ((size_t)b * Sv + s) * Dv) + tid;
        __builtin_nontemporal_store(r, dst);
    }
}

// Side output: embedding_EXON (B, S, 2) = [ 1.0 if 1<=x_1<=21 else 0.0 , EXON_POS ]
__global__ __launch_bounds__(256)
void embed_exon_kernel(const int*   __restrict__ x1,
                       const float* __restrict__ exon_pos, // (B, S, 1)
                       float*       __restrict__ out2)     // (B, S, 2)
{
    const int i = blockIdx.x * 256 + threadIdx.x;   // 0 .. B*S-1
    const int v = x1[i];
    v2f r;
    r.x = (v >= 1 && v <= 21) ? 1.0f : 0.0f;
    r.y = exon_pos[i];
    __builtin_nontemporal_store(r, (v2f*)out2 + i);
}

extern "C" void kernel_launch(void* const* d_in, const int* in_sizes, int n_in,
                              void* d_out, int out_size, void* d_ws, size_t ws_size,
                              hipStream_t stream) {
    // setup_inputs() order:
    //   0: x               (B,S)   int32
    //   1: x_1             (B,S)   int32
    //   2: EXON_POS        (B,S,1) float32
    //   3: tok_embed_weight(V,D)   float32
    //   4: pe              (S,D)   float32
    //   5: amino_pos_probs (V,S)   float32
    //   6: dipeptide_table (V,V)   float32
    const int*   x      = (const int*)  d_in[0];
    const int*   x1     = (const int*)  d_in[1];
    const float* exon   = (const float*)d_in[2];
    const float* tok_w  = (const float*)d_in[3];
    const float* pe     = (const float*)d_in[4];
    const float* probs  = (const float*)d_in[5];
    const float* dip    = (const float*)d_in[6];

    float* out_embed = (float*)d_out;                              // (B,S,D)
    float* out_exon  = out_embed + (size_t)Bv * Sv * Dv;           // (B,S,2)

    embed_main_kernel<<<dim3(Sv), dim3(128), 0, stream>>>(x, tok_w, pe, probs, dip, out_embed);
    embed_exon_kernel<<<dim3((Bv * Sv) / 256), dim3(256), 0, stream>>>(x1, exon, out_exon);
}